// SegmentationModule_14130442404026
// MI455X (gfx1250) — compile-verified
//
#include <hip/hip_runtime.h>

// Problem constants (from reference): B=8, N=20000, F=4, C=128, K=256, O=K+1=257
#define BB 8
#define NN 20000
#define FF 4
#define CC 128
#define KK 256
#define OO 257
#define WROW ((FF + CC) * KK)   // 33792 floats per W row

typedef __attribute__((ext_vector_type(2))) float v2f;
typedef __attribute__((ext_vector_type(4))) float v4f;
typedef __attribute__((ext_vector_type(8))) float v8f;

// ---------------------------------------------------------------------------
// Kernel 1: Wp[f,o] = sum_k W[o, 132k + f]     (4 x 257 floats into ws)
// ---------------------------------------------------------------------------
__global__ void wp_kernel(const float* __restrict__ W, float* __restrict__ Wp) {
    const int f = blockIdx.x;    // 0..3
    const int o = threadIdx.x;   // 0..256
    const float* row = W + (size_t)o * WROW + f;
    float acc = 0.f;
#pragma unroll 8
    for (int k = 0; k < KK; ++k) acc += row[k * (FF + CC)];
    Wp[f * OO + o] = acc;
}

// ---------------------------------------------------------------------------
// Kernel 2: s[b,o] = bias[o] + sum_{k,c} proposals[b,c,k] * W[o, 132k+4+c]
// One workgroup per o; thread t handles k=t; LDS float atomics reduce 8 batches.
// ---------------------------------------------------------------------------
__global__ __launch_bounds__(256) void s_kernel(const float* __restrict__ prop,
                                                const float* __restrict__ W,
                                                const float* __restrict__ bias,
                                                float* __restrict__ s) {
    const int o = blockIdx.x;    // 0..256
    const int k = threadIdx.x;   // 0..255
    __shared__ float ssum[BB];
    if (threadIdx.x < BB) ssum[threadIdx.x] = 0.f;
    __syncthreads();

    const float* wrow = W + (size_t)o * WROW + k * (FF + CC) + FF;  // 128 floats
    float acc[BB];
#pragma unroll
    for (int b = 0; b < BB; ++b) acc[b] = 0.f;

    for (int c = 0; c < CC; ++c) {
        const float w = wrow[c];
        const float* pb = prop + (size_t)c * KK + k;   // proposals[b,c,k]
#pragma unroll
        for (int b = 0; b < BB; ++b)
            acc[b] += pb[(size_t)b * CC * KK] * w;     // coalesced across k=tid
    }
#pragma unroll
    for (int b = 0; b < BB; ++b) atomicAdd(&ssum[b], acc[b]);
    __syncthreads();
    if (threadIdx.x < BB) s[threadIdx.x * OO + o] = ssum[threadIdx.x] + bias[o];
}

// ---------------------------------------------------------------------------
// Kernel 3: main streaming pass with V_WMMA_F32_16X16X4_F32.
// out[b,o,n] = pc[b,n,:4] . Wp[:,o] + s[b,o] for o in 0..255.
// Wave layout: 16000 waves = 8(b) x 16(o-tile) x 125(n-groups of 10 tiles).
// ---------------------------------------------------------------------------
#define TPW 10          // n-tiles (of 16 points) per wave; 1250 = 125 * 10
#define NGROUPS 125

__global__ __launch_bounds__(256) void main_wmma(const float* __restrict__ pc,
                                                 const float* __restrict__ Wp,
                                                 const float* __restrict__ s,
                                                 float* __restrict__ out) {
    const int lane = threadIdx.x & 31;
    const int wave = blockIdx.x * (blockDim.x >> 5) + (threadIdx.x >> 5);
    const int ng = wave % NGROUPS;
    const int ot = (wave / NGROUPS) & 15;
    const int b  = wave / (NGROUPS * 16);

    const int col = lane & 15;          // o within tile / point within tile
    const int hi  = lane >> 4;          // 0: K=0,1 rows M=0..7 ; 1: K=2,3 rows M=8..15
    const int o   = ot * 16 + col;

    // B-matrix (4x16 f32, 2 VGPRs): lane holds column o; rows K = 2*hi, 2*hi+1.
    v2f bm;
    bm.x = Wp[(2 * hi + 0) * OO + o];
    bm.y = Wp[(2 * hi + 1) * OO + o];

    const float sv = s[b * OO + o];     // bias + proposal term, broadcast into C

    const float* pcb = pc + (size_t)b * NN * FF;
    float* orow = out + ((size_t)b * OO + o) * NN;   // this lane's output row

    int n0 = ng * (TPW * 16);
#pragma unroll
    for (int t = 0; t < TPW; ++t, n0 += 16) {
        // A-matrix (16x4 f32, 2 VGPRs): lane loads 8B of point (n0 + col);
        // lanes L and L+16 fetch adjacent halves -> 256B fully coalesced.
        const int p = n0 + col;
        v2f a = *(const v2f*)(pcb + (size_t)p * FF + hi * 2);

        v8f c;
#pragma unroll
        for (int i = 0; i < 8; ++i) c[i] = sv;

        // D = A x B + C   (v_wmma_f32_16x16x4_f32)
        c = __builtin_amdgcn_wmma_f32_16x16x4_f32(
                /*neg_a=*/false, a, /*neg_b=*/false, bm,
                /*c_mod=*/(short)0, c, /*reuse_a=*/false, /*reuse_b=*/false);

        // D layout: lane holds col o; c[r] = row M = r + 8*hi -> contiguous n.
        float* dst = orow + n0 + hi * 8;
        v4f lo = {c[0], c[1], c[2], c[3]};
        v4f hi4 = {c[4], c[5], c[6], c[7]};
        *(v4f*)(dst)     = lo;     // 16B-aligned: row base is 80000B-aligned
        *(v4f*)(dst + 4) = hi4;
    }
}

// ---------------------------------------------------------------------------
// Kernel 4: last output row o=256 (257 = 16*16 + 1), coalesced over n.
// ---------------------------------------------------------------------------
__global__ void last_row(const float* __restrict__ pc,
                         const float* __restrict__ Wp,
                         const float* __restrict__ s,
                         float* __restrict__ out) {
    const int n = blockIdx.x * blockDim.x + threadIdx.x;
    const int b = blockIdx.y;
    if (n >= NN) return;
    v4f p = *(const v4f*)(pc + ((size_t)b * NN + n) * FF);
    const float r = p.x * Wp[0 * OO + 256] + p.y * Wp[1 * OO + 256] +
                    p.z * Wp[2 * OO + 256] + p.w * Wp[3 * OO + 256] +
                    s[b * OO + 256];
    out[((size_t)b * OO + 256) * NN + n] = r;
}

// ---------------------------------------------------------------------------
extern "C" void kernel_launch(void* const* d_in, const int* in_sizes, int n_in,
                              void* d_out, int out_size, void* d_ws, size_t ws_size,
                              hipStream_t stream) {
    const float* pc   = (const float*)d_in[0];   // (8, 20000, 4)
    const float* prop = (const float*)d_in[1];   // (8, 128, 256)
    const float* W    = (const float*)d_in[2];   // (257, 33792)
    const float* bias = (const float*)d_in[3];   // (257,)
    float* out = (float*)d_out;                  // (8, 257, 20000)

    float* Wp = (float*)d_ws;                    // 4*257 floats
    float* s  = Wp + FF * OO;                    // 8*257 floats

    wp_kernel<<<FF, OO, 0, stream>>>(W, Wp);
    s_kernel<<<OO, 256, 0, stream>>>(prop, W, bias, s);

    // 16000 waves -> 2000 blocks of 256 threads (8 waves each)
    main_wmma<<<(BB * 16 * NGROUPS) / 8, 256, 0, stream>>>(pc, Wp, s, out);
    last_row<<<dim3((NN + 255) / 256, BB), 256, 0, stream>>>(pc, Wp, s, out);
}